// Vegas_27487790695100
// MI455X (gfx1250) — compile-verified
//
#include <hip/hip_runtime.h>

typedef __attribute__((ext_vector_type(4))) float v4f;
typedef int v4i __attribute__((vector_size(16)));

#define TDIM 8

#ifndef __has_builtin
#define __has_builtin(x) 0
#endif

#if __has_builtin(__builtin_amdgcn_global_load_async_to_lds_b128)
#define HAVE_ASYNC_LDS 1
#else
#define HAVE_ASYNC_LDS 0
#endif

#if __has_builtin(__builtin_amdgcn_s_wait_asynccnt)
#define WAIT_ASYNC0() __builtin_amdgcn_s_wait_asynccnt(0)
#else
#define WAIT_ASYNC0() asm volatile("s_wait_asynccnt 0" ::: "memory")
#endif

typedef v4i __attribute__((address_space(1))) *gbl_v4i_ptr;
typedef v4i __attribute__((address_space(3))) *lds_v4i_ptr;

// x = grid[d][iu] + (grid[d][iu+1]-grid[d][iu]) * du   (clamped at upper edge)
// log_jac = log( prod_d h_d ) + TDIM * log(ninc)
__global__ __launch_bounds__(256) void vegas_map_kernel(
    const float* __restrict__ u,      // [n, TDIM]
    const float* __restrict__ grid,   // [TDIM, ninc+1]
    float* __restrict__ x_out,        // [n, TDIM]
    float* __restrict__ lj_out,       // [n]
    int n, int ninc)
{
    extern __shared__ __align__(16) float sgrid[];   // TDIM*(ninc+1) floats
    const int stride1 = ninc + 1;
    const int tot     = TDIM * stride1;              // multiple of 8
    const int nchunk  = tot >> 2;                    // b128 chunks

#if HAVE_ASYNC_LDS
    for (int c = threadIdx.x; c < nchunk; c += blockDim.x) {
        __builtin_amdgcn_global_load_async_to_lds_b128(
            (gbl_v4i_ptr)(grid + 4 * c),
            (lds_v4i_ptr)(sgrid + 4 * c),
            0, 0);
    }
    WAIT_ASYNC0();
#else
    for (int c = threadIdx.x; c < nchunk; c += blockDim.x) {
        ((v4f*)sgrid)[c] = ((const v4f*)grid)[c];
    }
#endif
    __syncthreads();

    const float fninc       = (float)ninc;
    const float log_jac_add = (float)TDIM * __logf(fninc);

    const int gid     = blockIdx.x * blockDim.x + threadIdx.x;
    const int gstride = gridDim.x * blockDim.x;

    for (int i = gid; i < n; i += gstride) {
        const v4f* up = (const v4f*)(u + (size_t)i * TDIM);
        v4f u0 = __builtin_nontemporal_load(up + 0);
        v4f u1 = __builtin_nontemporal_load(up + 1);
        float uu[TDIM] = {u0.x, u0.y, u0.z, u0.w, u1.x, u1.y, u1.z, u1.w};

        float xx[TDIM];
        float prod_h = 1.0f;
#pragma unroll
        for (int d = 0; d < TDIM; ++d) {
            float t  = uu[d] * fninc;
            int   iu = (int)t;                 // t >= 0 -> trunc == floor
            iu = iu < 0 ? 0 : iu;
            const bool in = iu < ninc;
            const int  ic = in ? iu : (ninc - 1);
            float du = t - (float)iu;
            const float* row = sgrid + d * stride1 + ic;
            float g0 = row[0];                 // ds_load
            float g1 = row[1];                 // ds_load
            float h  = g1 - g0;                // == inc[d][ic] exactly
            xx[d]    = in ? fmaf(h, du, g0) : g1;
            prod_h  *= h;
        }
        float lj = __logf(prod_h) + log_jac_add;

        v4f* xp = (v4f*)(x_out + (size_t)i * TDIM);
        v4f o0 = {xx[0], xx[1], xx[2], xx[3]};
        v4f o1 = {xx[4], xx[5], xx[6], xx[7]};
        __builtin_nontemporal_store(o0, xp + 0);
        __builtin_nontemporal_store(o1, xp + 1);
        __builtin_nontemporal_store(lj, lj_out + i);
    }
}

extern "C" void kernel_launch(void* const* d_in, const int* in_sizes, int n_in,
                              void* d_out, int out_size, void* d_ws, size_t ws_size,
                              hipStream_t stream)
{
    (void)n_in; (void)out_size; (void)d_ws; (void)ws_size;

    const float* u    = (const float*)d_in[0];   // [n, 8]
    const float* grid = (const float*)d_in[1];   // [8, ninc+1]
    // d_in[2] (inc) not needed: inc == diff(grid); d_in[3] (ninc) derived below.

    const int ninc = in_sizes[2] / TDIM;
    const int n    = in_sizes[0] / TDIM;

    float* x_out  = (float*)d_out;
    float* lj_out = x_out + (size_t)n * TDIM;

    const int block  = 256;
    int       nblk   = (n + block - 1) / block;
    if (nblk > 3072) nblk = 3072;   // grid-stride: amortize 32KB LDS table fill

    const size_t shmem = (size_t)TDIM * (size_t)(ninc + 1) * sizeof(float);

    vegas_map_kernel<<<dim3(nblk), dim3(block), shmem, stream>>>(
        u, grid, x_out, lj_out, n, ninc);
}